// GCN_RES_65747359367440
// MI455X (gfx1250) — compile-verified
//
#include <hip/hip_runtime.h>
#include <hip/hip_bf16.h>

#define NNODES 50000
#define NEDGES 640000
#define DH 128
#define DOUT 64
#define LN_EPS 1e-5f

typedef float v2f __attribute__((ext_vector_type(2)));
typedef float v8f __attribute__((ext_vector_type(8)));

// ---------------------------------------------------------------------------
// Normalization precompute: deg[d] = sum_e w[e] (dst==d) + 1 (self loop),
// dinv = rsqrt(deg), norm[e] = dinv[src]*w*dinv[dst]
// ---------------------------------------------------------------------------
__global__ void gcn_deg_kernel(const int* __restrict__ dst, const float* __restrict__ w,
                               float* __restrict__ deg, int E) {
    int e = blockIdx.x * blockDim.x + threadIdx.x;
    if (e < E) atomicAdd(&deg[dst[e]], w[e]);
}

__global__ void gcn_dinv_kernel(float* __restrict__ deg, int n) {
    int i = blockIdx.x * blockDim.x + threadIdx.x;
    if (i < n) {
        float d = deg[i] + 1.0f;   // self-loop weight 1 => deg always > 0
        deg[i] = rsqrtf(d);
    }
}

__global__ void gcn_norm_kernel(const int* __restrict__ src, const int* __restrict__ dst,
                                const float* __restrict__ w, const float* __restrict__ dinv,
                                float* __restrict__ norm, int E) {
    int e = blockIdx.x * blockDim.x + threadIdx.x;
    if (e < E) norm[e] = dinv[src[e]] * w[e] * dinv[dst[e]];
}

// ---------------------------------------------------------------------------
// WMMA GEMM: out[N x ncols] = A[N x 128] @ W[ncols x 128]^T   (fp32 WMMA, K=4)
// One wave per 16x16 output tile. nrows must be a multiple of 16 (50000 = 3125*16).
// ---------------------------------------------------------------------------
__global__ void gcn_gemm_wmma(const float* __restrict__ A,
                              const float* __restrict__ W,
                              float* __restrict__ out,
                              int nrows, int ncols) {
    const int lane = threadIdx.x & 31;
    const int wid  = blockIdx.x * (blockDim.x >> 5) + (threadIdx.x >> 5);
    const int ctiles = ncols >> 4;
    const int rtiles = nrows >> 4;
    if (wid >= rtiles * ctiles) return;      // wave-uniform exit: EXEC stays all-ones
    const int rt = wid / ctiles;
    const int ct = wid - rt * ctiles;
    const int row0 = rt << 4;
    const int j0   = ct << 4;

    const int m  = lane & 15;
    const int ko = (lane >> 4) << 1;         // lanes 0-15: K pair {k,k+1}; 16-31: {k+2,k+3}
    const float* __restrict__ arow = A + (size_t)(row0 + m) * DH + ko;
    const float* __restrict__ brow = W + (size_t)(j0 + m) * DH + ko;

    v8f acc = {};
#pragma unroll
    for (int k = 0; k < DH; k += 4) {
        v2f a = *(const v2f*)(arow + k);
        v2f b = *(const v2f*)(brow + k);
        acc = __builtin_amdgcn_wmma_f32_16x16x4_f32(
            /*neg_a=*/false, a, /*neg_b=*/false, b,
            /*c_mod=*/(short)0, acc, /*reuse_a=*/false, /*reuse_b=*/false);
    }

    // C/D layout: VGPR v, lane L -> M = v + 8*(L>>4), N = L&15
    float* __restrict__ o = out + (size_t)(row0 + ((lane >> 4) << 3)) * ncols + j0 + m;
#pragma unroll
    for (int v = 0; v < 8; ++v)
        o[(size_t)v * ncols] = acc[v];
}

// ---------------------------------------------------------------------------
// Edge scatter: agg[dst] += norm[e] * xw[src]. One thread = one edge x 4 feats.
// ---------------------------------------------------------------------------
__global__ void gcn_scatter_kernel(const float* __restrict__ xw,
                                   const int* __restrict__ src, const int* __restrict__ dst,
                                   const float* __restrict__ norm,
                                   float* __restrict__ agg, int E, int ncols) {
    int t = blockIdx.x * blockDim.x + threadIdx.x;
    int per_edge = ncols >> 2;
    int e = t / per_edge;
    if (e >= E) return;
    int f = (t - e * per_edge) << 2;
    int s = src[e], d = dst[e];
    float w = norm[e];
    const float4 v = *(const float4*)(xw + (size_t)s * ncols + f);
    float* p = agg + (size_t)d * ncols + f;
    atomicAdd(p + 0, w * v.x);
    atomicAdd(p + 1, w * v.y);
    atomicAdd(p + 2, w * v.z);
    atomicAdd(p + 3, w * v.w);
}

// ---------------------------------------------------------------------------
// Wave32 reduction
// ---------------------------------------------------------------------------
__device__ __forceinline__ float wave_sum32(float v) {
#pragma unroll
    for (int m = 16; m >= 1; m >>= 1) v += __shfl_xor(v, m, 32);
    return v;
}

// ---------------------------------------------------------------------------
// Hidden epilogue: h = relu(LN(agg + dinv^2*xw + b)) + res   (one wave / node)
// Folds the self-loop edge (norm = dinv^2, src == dst) in here.
// ---------------------------------------------------------------------------
__global__ void gcn_epilogue_hidden(const float* __restrict__ agg, const float* __restrict__ xw,
                                    const float* __restrict__ dinv,
                                    const float* __restrict__ bias, const float* __restrict__ gamma,
                                    const float* __restrict__ beta,
                                    const float* __restrict__ res, float* __restrict__ hout, int n) {
    int t = blockIdx.x * blockDim.x + threadIdx.x;
    int lane = t & 31;
    int node = t >> 5;
    if (node >= n) return;

    float di = dinv[node];
    float sl = di * di;
    size_t base = (size_t)node * DH + (lane << 2);

    float4 a  = *(const float4*)(agg + base);
    float4 xv = *(const float4*)(xw + base);
    float4 b  = *(const float4*)(bias + (lane << 2));

    float v0 = fmaf(sl, xv.x, a.x) + b.x;
    float v1 = fmaf(sl, xv.y, a.y) + b.y;
    float v2 = fmaf(sl, xv.z, a.z) + b.z;
    float v3 = fmaf(sl, xv.w, a.w) + b.w;

    float s  = wave_sum32(v0 + v1 + v2 + v3);
    float sq = wave_sum32(v0 * v0 + v1 * v1 + v2 * v2 + v3 * v3);
    float mean = s * (1.0f / DH);
    float var  = sq * (1.0f / DH) - mean * mean;
    float rinv = rsqrtf(var + LN_EPS);

    float4 g  = *(const float4*)(gamma + (lane << 2));
    float4 be = *(const float4*)(beta + (lane << 2));
    float4 r  = *(const float4*)(res + base);

    float4 o;
    o.x = fmaxf(g.x * (v0 - mean) * rinv + be.x, 0.0f) + r.x;
    o.y = fmaxf(g.y * (v1 - mean) * rinv + be.y, 0.0f) + r.y;
    o.z = fmaxf(g.z * (v2 - mean) * rinv + be.z, 0.0f) + r.z;
    o.w = fmaxf(g.w * (v3 - mean) * rinv + be.w, 0.0f) + r.w;
    *(float4*)(hout + base) = o;
}

// ---------------------------------------------------------------------------
// Final epilogue: out = LN(agg + dinv^2*xw + b)  over 64 features, no relu/res
// ---------------------------------------------------------------------------
__global__ void gcn_epilogue_final(const float* __restrict__ agg, const float* __restrict__ xw,
                                   const float* __restrict__ dinv,
                                   const float* __restrict__ bias, const float* __restrict__ gamma,
                                   const float* __restrict__ beta,
                                   float* __restrict__ out, int n) {
    int t = blockIdx.x * blockDim.x + threadIdx.x;
    int lane = t & 31;
    int node = t >> 5;
    if (node >= n) return;

    float di = dinv[node];
    float sl = di * di;
    size_t base = (size_t)node * DOUT + (lane << 1);

    float2 a  = *(const float2*)(agg + base);
    float2 xv = *(const float2*)(xw + base);
    float2 b  = *(const float2*)(bias + (lane << 1));

    float v0 = fmaf(sl, xv.x, a.x) + b.x;
    float v1 = fmaf(sl, xv.y, a.y) + b.y;

    float s  = wave_sum32(v0 + v1);
    float sq = wave_sum32(v0 * v0 + v1 * v1);
    float mean = s * (1.0f / DOUT);
    float var  = sq * (1.0f / DOUT) - mean * mean;
    float rinv = rsqrtf(var + LN_EPS);

    float2 g  = *(const float2*)(gamma + (lane << 1));
    float2 be = *(const float2*)(beta + (lane << 1));

    float2 o;
    o.x = g.x * (v0 - mean) * rinv + be.x;
    o.y = g.y * (v1 - mean) * rinv + be.y;
    *(float2*)(out + base) = o;
}

// ---------------------------------------------------------------------------
extern "C" void kernel_launch(void* const* d_in, const int* in_sizes, int n_in,
                              void* d_out, int out_size, void* d_ws, size_t ws_size,
                              hipStream_t stream) {
    const float* x    = (const float*)d_in[0];    // [N,128]
    const int*   ei   = (const int*)d_in[1];      // [2,E] -> src = ei, dst = ei+E
    const float* ew   = (const float*)d_in[2];    // [E]
    const float* W03  = (const float*)d_in[3];    // [4,128,128]
    const float* b03  = (const float*)d_in[4];    // [4,128]
    const float* g03  = (const float*)d_in[5];    // [4,128]
    const float* be03 = (const float*)d_in[6];    // [4,128]
    const float* W4   = (const float*)d_in[7];    // [64,128]
    const float* b4   = (const float*)d_in[8];
    const float* g4   = (const float*)d_in[9];
    const float* be4  = (const float*)d_in[10];

    const int* srcI = ei;
    const int* dstI = ei + NEDGES;

    float* dinv = (float*)d_ws;                       // N
    float* norm = dinv + NNODES;                      // E
    float* xw   = norm + NEDGES;                      // N*128
    float* agg  = xw + (size_t)NNODES * DH;           // N*128
    float* hbuf = agg + (size_t)NNODES * DH;          // N*128

    // --- degree / normalization ---
    hipMemsetAsync(dinv, 0, (size_t)NNODES * sizeof(float), stream);
    gcn_deg_kernel<<<(NEDGES + 255) / 256, 256, 0, stream>>>(dstI, ew, dinv, NEDGES);
    gcn_dinv_kernel<<<(NNODES + 255) / 256, 256, 0, stream>>>(dinv, NNODES);
    gcn_norm_kernel<<<(NEDGES + 255) / 256, 256, 0, stream>>>(srcI, dstI, ew, dinv, norm, NEDGES);

    // --- 4 hidden GCN layers ---
    const int rtiles = NNODES / 16;                   // 3125
    const float* hin = x;
    const float* resin = x;
    for (int l = 0; l < 4; ++l) {
        int waves  = rtiles * (DH / 16);              // 25000
        int blocks = (waves * 32 + 255) / 256;        // 3125
        gcn_gemm_wmma<<<blocks, 256, 0, stream>>>(hin, W03 + (size_t)l * DH * DH, xw, NNODES, DH);

        hipMemsetAsync(agg, 0, (size_t)NNODES * DH * sizeof(float), stream);
        long long sthreads = (long long)NEDGES * (DH / 4);
        gcn_scatter_kernel<<<(int)((sthreads + 255) / 256), 256, 0, stream>>>(
            xw, srcI, dstI, norm, agg, NEDGES, DH);

        long long ethreads = (long long)NNODES * 32;
        gcn_epilogue_hidden<<<(int)((ethreads + 255) / 256), 256, 0, stream>>>(
            agg, xw, dinv, b03 + l * DH, g03 + l * DH, be03 + l * DH, resin, hbuf, NNODES);

        hin = hbuf;
        resin = hbuf;
    }

    // --- final layer (128 -> 64), LayerNorm only ---
    {
        int waves  = rtiles * (DOUT / 16);            // 12500
        int blocks = (waves * 32 + 255) / 256;        // 1563 (guarded)
        gcn_gemm_wmma<<<blocks, 256, 0, stream>>>(hbuf, W4, xw, NNODES, DOUT);

        hipMemsetAsync(agg, 0, (size_t)NNODES * DOUT * sizeof(float), stream);
        long long sthreads = (long long)NEDGES * (DOUT / 4);
        gcn_scatter_kernel<<<(int)((sthreads + 255) / 256), 256, 0, stream>>>(
            xw, srcI, dstI, norm, agg, NEDGES, DOUT);

        long long ethreads = (long long)NNODES * 32;
        gcn_epilogue_final<<<(int)((ethreads + 255) / 256), 256, 0, stream>>>(
            agg, xw, dinv, b4, g4, be4, (float*)d_out, NNODES);
    }
}